// GuidedFilter_82566451298663
// MI455X (gfx1250) — compile-verified
//
#include <hip/hip_runtime.h>

typedef float v2f __attribute__((ext_vector_type(2)));
typedef float v4f __attribute__((ext_vector_type(4)));
typedef float v8f __attribute__((ext_vector_type(8)));
typedef int   v4i __attribute__((vector_size(16)));

static __device__ __forceinline__ int imin(int a, int b) { return a < b ? a : b; }
static __device__ __forceinline__ int imax(int a, int b) { return a > b ? a : b; }

// D = A(16x4) * B(4x16) + C, fp32 WMMA:
// (neg_a, A, neg_b, B, c_mod, C, reuse_a, reuse_b)
#define WMMA4(a, b, c) \
  __builtin_amdgcn_wmma_f32_16x16x4_f32(false, (a), false, (b), (short)0, (c), false, false)

// ---- CDNA5 async global->LDS copy (guarded; fallback = vec4 through VGPRs) ----
#if defined(__has_builtin)
#if __has_builtin(__builtin_amdgcn_global_load_async_to_lds_b128)
#define HAVE_ASYNC_LDS 1
#endif
#if __has_builtin(__builtin_amdgcn_s_wait_asynccnt)
#define HAVE_WAIT_ASYNC 1
#endif
#endif

typedef __attribute__((address_space(1))) v4i gv4i;   // global int4
typedef __attribute__((address_space(3))) v4i lv4i;   // LDS int4
static __device__ __forceinline__ gv4i* to_gbl(const void* p) {
  return (gv4i*)(unsigned long long)p;                 // strips const via int cast
}
static __device__ __forceinline__ lv4i* to_lds(void* p) {
  return (lv4i*)(unsigned int)(unsigned long long)p;   // AS3 ptr = low 32 bits
}

#define IMG    512
#define TILE   64
#define LOADN  96          // TILE + 2*16 halo (two box passes, R=8 each)
#define MIDN   80          // TILE + 2*8
#define SX_ST  100         // 16B-aligned rows for async b128; conflict-free columns
#define H1_ST  81
#define AB_ST  81

#define SX_OFF   0
#define H1X_OFF  (LOADN * SX_ST)             // 96x100 x-region
#define H1XX_OFF (H1X_OFF + LOADN * H1_ST)   // 96x81 H(x)   (reused for H2(A))
#define ABA_OFF  (H1XX_OFF + LOADN * H1_ST)  // 96x81 H(x^2) (reused for H2(b))
#define ABB_OFF  (ABA_OFF + MIDN * AB_ST)    // 80x81 A
#define LDS_N    (ABB_OFF + MIDN * AB_ST)    // 80x81 b  -> 38112 floats = 152448 B

__global__ void __launch_bounds__(256)
guided_filter_wmma(const float* __restrict__ x, float* __restrict__ out) {
  __shared__ float lds[LDS_N];
  float* SX   = lds + SX_OFF;
  float* H1X  = lds + H1X_OFF;   // phase1: H(x);  phase3: H2(A)
  float* H1XX = lds + H1XX_OFF;  // phase1: H(xx); phase3: H2(b)
  float* ABA  = lds + ABA_OFF;
  float* ABB  = lds + ABB_OFF;

  const int tid   = threadIdx.x;
  const int lane  = tid & 31;
  const int wave  = tid >> 5;
  const int i16   = lane & 15;   // M-row (A) / N-col (B,C,D) within 16
  const int hi    = lane >> 4;   // half-wave
  const int khalf = hi << 1;     // K offset of this half's A/B elements

  const int tx = blockIdx.x, ty = blockIdx.y, plane = blockIdx.z;
  const int gx0 = tx * TILE - 16, gy0 = ty * TILE - 16;
  const long long poff = (long long)plane * (IMG * IMG);

  // ---- Phase 0: load 96x96 x region, zero-padded (clamped box == padded box) ----
  const bool interior = (gy0 >= 0) && (gy0 + LOADN <= IMG) &&
                        (gx0 >= 0) && (gx0 + LOADN <= IMG);
  if (interior) {
    const float* src = x + poff + (long long)gy0 * IMG + gx0;   // 64B aligned
    for (int idx = tid; idx < LOADN * (LOADN / 4); idx += 256) { // 2304 x 16B
      int r  = idx / (LOADN / 4);
      int c4 = idx - r * (LOADN / 4);
#if defined(HAVE_ASYNC_LDS)
      __builtin_amdgcn_global_load_async_to_lds_b128(
          to_gbl(src + r * IMG + (c4 << 2)),
          to_lds(SX + r * SX_ST + (c4 << 2)), 0, 0);
#else
      *(v4f*)(SX + r * SX_ST + (c4 << 2)) = *(const v4f*)(src + r * IMG + (c4 << 2));
#endif
    }
#if defined(HAVE_ASYNC_LDS)
#if defined(HAVE_WAIT_ASYNC)
    __builtin_amdgcn_s_wait_asynccnt(0);
#else
    asm volatile("s_wait_asynccnt 0x0" ::: "memory");
#endif
#endif
  } else {
    for (int idx = tid; idx < LOADN * LOADN; idx += 256) {
      int r = idx / LOADN;
      int c = idx - r * LOADN;
      int gy = gy0 + r, gx = gx0 + c;
      float v = 0.0f;
      if ((unsigned)gy < (unsigned)IMG && (unsigned)gx < (unsigned)IMG)
        v = x[poff + gy * IMG + gx];
      SX[r * SX_ST + c] = v;
    }
  }
  __syncthreads();

  // ---- Phase 1: horizontal sums H = X * Band (96 rows x 80 cols): x then x^2 ----
  // H[r][C0+j] = sum_{k=j..j+16} X[r][C0+k]
  for (int t = wave; t < 30; t += 8) {                  // 6x5 tiles, H(x)
    int tr = t / 5, tc = t - (t / 5) * 5;
    int R0 = tr * 16, C0 = tc * 16;
    v8f acc = {0.f, 0.f, 0.f, 0.f, 0.f, 0.f, 0.f, 0.f};
#pragma unroll
    for (int kk = 0; kk < 8; ++kk) {
      int k0 = kk * 4 + khalf;
      v2f a, b;
      a.x = SX[(R0 + i16) * SX_ST + C0 + k0];
      a.y = SX[(R0 + i16) * SX_ST + C0 + k0 + 1];
      b.x = ((unsigned)(k0     - i16) <= 16u) ? 1.0f : 0.0f;  // Band[k][j]=j<=k<=j+16
      b.y = ((unsigned)(k0 + 1 - i16) <= 16u) ? 1.0f : 0.0f;
      acc = WMMA4(a, b, acc);
    }
#pragma unroll
    for (int v = 0; v < 8; ++v)
      H1X[(R0 + v + (hi << 3)) * H1_ST + C0 + i16] = acc[v];
  }
  for (int t = wave; t < 30; t += 8) {                  // 6x5 tiles, H(x^2)
    int tr = t / 5, tc = t - (t / 5) * 5;
    int R0 = tr * 16, C0 = tc * 16;
    v8f acc = {0.f, 0.f, 0.f, 0.f, 0.f, 0.f, 0.f, 0.f};
#pragma unroll
    for (int kk = 0; kk < 8; ++kk) {
      int k0 = kk * 4 + khalf;
      v2f a, b;
      float s0 = SX[(R0 + i16) * SX_ST + C0 + k0];
      float s1 = SX[(R0 + i16) * SX_ST + C0 + k0 + 1];
      a.x = s0 * s0;
      a.y = s1 * s1;
      b.x = ((unsigned)(k0     - i16) <= 16u) ? 1.0f : 0.0f;
      b.y = ((unsigned)(k0 + 1 - i16) <= 16u) ? 1.0f : 0.0f;
      acc = WMMA4(a, b, acc);
    }
#pragma unroll
    for (int v = 0; v < 8; ++v)
      H1XX[(R0 + v + (hi << 3)) * H1_ST + C0 + i16] = acc[v];
  }
  __syncthreads();

  // ---- Phase 2: vertical sums V = Band * H (80x80), then pointwise A,b ----
  for (int job = wave; job < 25; job += 8) {            // 5x5 tiles
    int tr = job / 5, tc = job - (job / 5) * 5;
    int R0 = tr * 16, C0 = tc * 16;
    v8f ax  = {0.f, 0.f, 0.f, 0.f, 0.f, 0.f, 0.f, 0.f};
    v8f axx = {0.f, 0.f, 0.f, 0.f, 0.f, 0.f, 0.f, 0.f};
#pragma unroll
    for (int kk = 0; kk < 8; ++kk) {
      int k0 = kk * 4 + khalf;
      v2f a, bx, bxx;
      a.x = ((unsigned)(k0     - i16) <= 16u) ? 1.0f : 0.0f;  // Band[i][k]=i<=k<=i+16
      a.y = ((unsigned)(k0 + 1 - i16) <= 16u) ? 1.0f : 0.0f;
      bx.x  = H1X [(R0 + k0    ) * H1_ST + C0 + i16];
      bx.y  = H1X [(R0 + k0 + 1) * H1_ST + C0 + i16];
      bxx.x = H1XX[(R0 + k0    ) * H1_ST + C0 + i16];
      bxx.y = H1XX[(R0 + k0 + 1) * H1_ST + C0 + i16];
      ax  = WMMA4(a, bx,  ax);
      axx = WMMA4(a, bxx, axx);
    }
#pragma unroll
    for (int v = 0; v < 8; ++v) {
      int row = R0 + v + (hi << 3);
      int col = C0 + i16;
      int gy = gy0 + 8 + row, gx = gx0 + 8 + col;
      int cy = imin(gy + 8, IMG - 1) - imax(gy - 8, 0) + 1;
      int cx = imin(gx + 8, IMG - 1) - imax(gx - 8, 0) + 1;
      float Ninv = 1.0f / (float)imax(cy * cx, 1);
      float mean = ax[v] * Ninv;
      float var  = axx[v] * Ninv - mean * mean;   // == cov_xy in the reference
      float Aval = var / (var + 1e-8f);
      float bval = mean - Aval * mean;
      bool inb = ((unsigned)gy < (unsigned)IMG) && ((unsigned)gx < (unsigned)IMG);
      if (!inb) { Aval = 0.0f; bval = 0.0f; }     // zero-pad A,b for stage-2 sums
      ABA[row * AB_ST + col] = Aval;
      ABB[row * AB_ST + col] = bval;
    }
  }
  __syncthreads();

  // ---- Phase 3: horizontal sums of A,b (80 rows x 64 cols) into reused H buffers ----
  for (int t = wave; t < 20; t += 8) {                  // 5x4 tiles, H2(A)
    int tr = t / 4, tc = t - (t / 4) * 4;
    int R0 = tr * 16, C0 = tc * 16;
    v8f acc = {0.f, 0.f, 0.f, 0.f, 0.f, 0.f, 0.f, 0.f};
#pragma unroll
    for (int kk = 0; kk < 8; ++kk) {
      int k0 = kk * 4 + khalf;
      v2f a, b;
      a.x = ABA[(R0 + i16) * AB_ST + C0 + k0];
      a.y = ABA[(R0 + i16) * AB_ST + C0 + k0 + 1];
      b.x = ((unsigned)(k0     - i16) <= 16u) ? 1.0f : 0.0f;
      b.y = ((unsigned)(k0 + 1 - i16) <= 16u) ? 1.0f : 0.0f;
      acc = WMMA4(a, b, acc);
    }
#pragma unroll
    for (int v = 0; v < 8; ++v)
      H1X[(R0 + v + (hi << 3)) * H1_ST + C0 + i16] = acc[v];
  }
  for (int t = wave; t < 20; t += 8) {                  // 5x4 tiles, H2(b)
    int tr = t / 4, tc = t - (t / 4) * 4;
    int R0 = tr * 16, C0 = tc * 16;
    v8f acc = {0.f, 0.f, 0.f, 0.f, 0.f, 0.f, 0.f, 0.f};
#pragma unroll
    for (int kk = 0; kk < 8; ++kk) {
      int k0 = kk * 4 + khalf;
      v2f a, b;
      a.x = ABB[(R0 + i16) * AB_ST + C0 + k0];
      a.y = ABB[(R0 + i16) * AB_ST + C0 + k0 + 1];
      b.x = ((unsigned)(k0     - i16) <= 16u) ? 1.0f : 0.0f;
      b.y = ((unsigned)(k0 + 1 - i16) <= 16u) ? 1.0f : 0.0f;
      acc = WMMA4(a, b, acc);
    }
#pragma unroll
    for (int v = 0; v < 8; ++v)
      H1XX[(R0 + v + (hi << 3)) * H1_ST + C0 + i16] = acc[v];
  }
  __syncthreads();

  // ---- Phase 4: vertical sums of H2(A),H2(b) (64x64) + final output ----
  for (int job = wave; job < 16; job += 8) {            // 4x4 tiles
    int tr = job >> 2, tc = job & 3;
    int R0 = tr * 16, C0 = tc * 16;
    v8f aA = {0.f, 0.f, 0.f, 0.f, 0.f, 0.f, 0.f, 0.f};
    v8f aB = {0.f, 0.f, 0.f, 0.f, 0.f, 0.f, 0.f, 0.f};
#pragma unroll
    for (int kk = 0; kk < 8; ++kk) {
      int k0 = kk * 4 + khalf;
      v2f a, bA, bB;
      a.x = ((unsigned)(k0     - i16) <= 16u) ? 1.0f : 0.0f;
      a.y = ((unsigned)(k0 + 1 - i16) <= 16u) ? 1.0f : 0.0f;
      bA.x = H1X [(R0 + k0    ) * H1_ST + C0 + i16];
      bA.y = H1X [(R0 + k0 + 1) * H1_ST + C0 + i16];
      bB.x = H1XX[(R0 + k0    ) * H1_ST + C0 + i16];
      bB.y = H1XX[(R0 + k0 + 1) * H1_ST + C0 + i16];
      aA = WMMA4(a, bA, aA);
      aB = WMMA4(a, bB, aB);
    }
#pragma unroll
    for (int v = 0; v < 8; ++v) {
      int row = R0 + v + (hi << 3);
      int col = C0 + i16;
      int gy = ty * TILE + row, gx = tx * TILE + col;
      int cy = imin(gy + 8, IMG - 1) - imax(gy - 8, 0) + 1;
      int cx = imin(gx + 8, IMG - 1) - imax(gx - 8, 0) + 1;
      float Ninv = 1.0f / (float)(cy * cx);
      float xv = SX[(row + 16) * SX_ST + (col + 16)];
      out[poff + gy * IMG + gx] = aA[v] * Ninv * xv + aB[v] * Ninv;
    }
  }
}

extern "C" void kernel_launch(void* const* d_in, const int* in_sizes, int n_in,
                              void* d_out, int out_size, void* d_ws, size_t ws_size,
                              hipStream_t stream) {
  const float* xin = (const float*)d_in[0];   // 'ref' (d_in[1]) is unused by the reference math
  float* out = (float*)d_out;
  int planes = in_sizes[0] / (IMG * IMG);     // n*t*c = 96
  dim3 grid(IMG / TILE, IMG / TILE, planes);  // 8 x 8 x 96
  dim3 block(256);                            // 8 wave32
  hipLaunchKernelGGL(guided_filter_wmma, grid, block, 0, stream, xin, out);
}